// GCNext_21723944583549
// MI455X (gfx1250) — compile-verified
//
#include <hip/hip_runtime.h>
#include <hip/hip_bf16.h>

// Fully fused GCNext forward for MI455X (gfx1250, wave32).
// One workgroup (256 threads = 8 waves) per batch sample; h[66][50] lives in
// LDS across all 24 layers. Dense matmuls use V_WMMA_F32_16X16X4_F32 with
// branch-free fragment loads (row/col clamping + compile-time K edge).

#define TT   50      // T (frames)
#define VD   66      // V (features) = 22 joints * 3
#define NL   24      // layers
#define NJ   22
#define HP   52      // LDS pitch for [V][T]-ish tiles
#define XP   68      // LDS pitch for [T][V]-ish tiles
#define NWAVES 8

#define SZ_H   (80 * HP)          // C-store padding: 5 M-tiles of 16 rows
#define SZ_SA  (80 * HP)
#define SZ_SB  (64 * XP)          // max(64*68, 80*52)

typedef __attribute__((ext_vector_type(2))) float v2f;
typedef __attribute__((ext_vector_type(8))) float v8f;

// C[M,N] = op(A)[M,K] * op(B)[K,N] (+ bias[n]); 16x16 tiles, K-step 4.
// TA: A(m,k)=A[k*lda+m] else A[m*lda+k].  TB: B(k,n)=B[n*ldb+k] else B[k*ldb+n].
// A-row / B-col indices are CLAMPED (never OOB, guards are pure selects done
// once per tile); the K edge is handled only in the statically-last unrolled
// iteration. GM: guard C stores on m<M (needed only for the global C).
// EXEC is full at every WMMA.
template <int M, int K, int N, bool TA, bool TB, bool GM>
__device__ __forceinline__ void wmma_f32_gemm(
    const float* __restrict__ A, int lda,
    const float* __restrict__ B, int ldb,
    float* __restrict__ C, int ldc,
    const float* __restrict__ bias)
{
    constexpr int tilesM = (M + 15) / 16;
    constexpr int tilesN = (N + 15) / 16;
    const int tid  = threadIdx.x;
    const int wave = tid >> 5;
    const int lane = tid & 31;
    const int l16  = lane & 15;
    const int kh   = (lane >> 4) << 1;          // 0 or 2 (ISA A/B k-half)

    for (int tile = wave; tile < tilesM * tilesN; tile += NWAVES) {
        const int m0 = (tile / tilesN) << 4;
        const int n0 = (tile % tilesN) << 4;
        const int am = m0 + l16;
        const int bn = n0 + l16;
        const int ar = (am < M) ? am : (M - 1);  // clamped: always in-bounds
        const int br = (bn < N) ? bn : (N - 1);  // (dup rows land in discarded C)
        v8f acc = {0.f, 0.f, 0.f, 0.f, 0.f, 0.f, 0.f, 0.f};

#pragma unroll
        for (int k0 = 0; k0 < K; k0 += 4) {
            v2f a, b;
            if (k0 + 3 < K) {                    // constant per unrolled iter
                const int ka = k0 + kh;
                if (TA) { a.x = A[ka * lda + ar]; a.y = A[(ka + 1) * lda + ar]; }
                else    { a = *(const v2f*)(A + ar * lda + ka); }     // ds/global b64
                if (TB) { b = *(const v2f*)(B + br * ldb + ka); }     // ds/global b64
                else    { b.x = B[ka * ldb + br]; b.y = B[(ka + 1) * ldb + br]; }
            } else {                             // K edge: zero-fill via selects
                const int ka = k0 + kh;
                const bool e0 = (ka     < K);
                const bool e1 = (ka + 1 < K);
                const int k1 = e0 ? ka     : 0;
                const int k2 = e1 ? ka + 1 : 0;
                a.x = TA ? A[k1 * lda + ar] : A[ar * lda + k1];
                a.y = TA ? A[k2 * lda + ar] : A[ar * lda + k2];
                b.x = TB ? B[br * ldb + k1] : B[k1 * ldb + br];
                b.y = TB ? B[br * ldb + k2] : B[k2 * ldb + br];
                if (!e0) { a.x = 0.f; b.x = 0.f; }
                if (!e1) { a.y = 0.f; b.y = 0.f; }
            }
            acc = __builtin_amdgcn_wmma_f32_16x16x4_f32(
                false, a, false, b, (short)0, acc, false, false);
        }

        // C/D layout: vgpr r -> M = m0 + r + 8*(lane>=16), N = n0 + (lane&15)
        const float bv    = bias ? bias[br] : 0.0f;
        const int   mBase = m0 + ((lane >> 4) << 3);
        if (bn < N) {
#pragma unroll
            for (int r = 0; r < 8; ++r) {
                const int m = mBase + r;
                if (!GM || m < M) C[m * ldc + bn] = acc[r] + bv;
            }
        }
    }
}

__global__ void __launch_bounds__(256)
gcnext_fused(const float* __restrict__ x,
             const float* __restrict__ fc_in_w, const float* __restrict__ fc_in_b,
             const float* __restrict__ in_w,
             const float* __restrict__ adj_j,  const float* __restrict__ adj_t,
             const float* __restrict__ adj_jc, const float* __restrict__ adj_tj,
             const float* __restrict__ upd_w,  const float* __restrict__ upd_b,
             const float* __restrict__ ln_a,   const float* __restrict__ ln_b,
             const float* __restrict__ mlp,
             const float* __restrict__ out_w,
             const float* __restrict__ fc_out_w, const float* __restrict__ fc_out_b,
             const float* __restrict__ gumbel,
             float* __restrict__ out, int Bn)
{
    __shared__ __align__(16) float h  [SZ_H];    // resident state h[v][t]
    __shared__ __align__(16) float sA [SZ_SA];   // x-stage / (x1+mix) / g
    __shared__ __align__(16) float sB [SZ_SB];   // fc_in tmp / y
    __shared__ __align__(16) float wbuf[TT * HP];// staged upd_w[f][t]
    __shared__ __align__(16) float ajs[NJ * NJ]; // staged adj_j
    __shared__ float hbar[TT];
    __shared__ float mu[TT], rstd[TT];
    __shared__ int   gsel;

    const int b   = blockIdx.x;
    const int tid = threadIdx.x;

    // ---- stage x[b] : [T,V] into sA with pitch XP (coalesced) ----
    for (int i = tid; i < TT * VD; i += 256)
        sA[(i / VD) * XP + (i % VD)] = x[b * TT * VD + i];
    __syncthreads();

    // ---- tmp[t][v] = x @ fc_in_w^T + b  (M=T,K=V,N=V) -> sB pitch XP ----
    wmma_f32_gemm<TT, VD, VD, false, true, false>(sA, XP, fc_in_w, VD, sB, XP, fc_in_b);
    __syncthreads();

    // ---- h[v][t] = tmp^T @ in_weight  (M=V,K=T,N=T) ----
    wmma_f32_gemm<VD, TT, TT, true, false, false>(sB, XP, in_w, TT, h, HP, nullptr);
    __syncthreads();

#pragma unroll 1
    for (int l = 0; l < NL; ++l) {
        const float* aj  = adj_j  + l * NJ * NJ;
        const float* at  = adj_t  + l * TT * TT;
        const float* ajc = adj_jc + l * NJ * 9;
        const float* atj = adj_tj + l * VD * TT * TT;
        const float* uw  = upd_w  + l * TT * TT;
        const float* ub  = upd_b  + l * TT;
        const float* la  = ln_a   + l * VD;
        const float* lb  = ln_b   + l * VD;

        // stage adjacency + update weight into LDS
        for (int i = tid; i < NJ * NJ; i += 256) ajs[i] = aj[i];
        for (int i = tid; i < TT * TT; i += 256)
            wbuf[(i / TT) * HP + (i % TT)] = uw[i];
        // speculative prefetch of next layer's update weight into GL2
        if (l + 1 < NL) {
            const float* nuw = upd_w + (l + 1) * TT * TT;
            for (int i = tid; i < 79; i += 256)
                __builtin_prefetch(nuw + i * 32, 0, 1);
        }
        __syncthreads();

        // ---- routing gate: argmax(h.mean(V) @ mlp + gumbel) ----
        for (int t = tid; t < TT; t += 256) {
            float s = 0.f;
            for (int v = 0; v < VD; ++v) s += h[v * HP + t];
            hbar[t] = s * (1.0f / (float)VD);
        }
        __syncthreads();
        if (tid == 0) {
            const float* g = gumbel + ((size_t)l * Bn + b) * 4;
            float best = -3.4e38f; int bi = 0;
            for (int k = 0; k < 4; ++k) {
                float p = 0.f;
                for (int t = 0; t < TT; ++t) p += hbar[t] * mlp[t * 4 + k];
                p += g[k];
                if (p > best) { best = p; bi = k; }   // first-max tie rule (argmax)
            }
            gsel = bi;
        }
        __syncthreads();
        const int gs = gsel;   // uniform across block -> no divergence

        // ---- sA[v][t] = x1 (joint adj) + selected mix branch ----
        for (int i = tid; i < VD * TT; i += 256) {
            const int v = i / TT, t = i % TT;
            const int j = v / 3, c = v % 3;
            float acc = 0.f;
#pragma unroll
            for (int jp = 0; jp < NJ; ++jp)
                acc += ajs[j * NJ + jp] * h[(jp * 3 + c) * HP + t];
            if (gs == 1) {                 // banded temporal adjacency
                if (t > 0)      acc += at[t * TT + t - 1] * h[v * HP + t - 1];
                if (t < TT - 1) acc += at[t * TT + t + 1] * h[v * HP + t + 1];
            } else if (gs == 2) {          // 3x3 joint-channel mix
                const float* a3 = ajc + j * 9 + c * 3;
                acc += a3[0] * h[(j * 3 + 0) * HP + t]
                     + a3[1] * h[(j * 3 + 1) * HP + t]
                     + a3[2] * h[(j * 3 + 2) * HP + t];
            } else if (gs == 3) {          // per-feature banded temporal adj
                const float* ar = atj + ((size_t)v * TT + t) * TT;
                if (t > 0)      acc += ar[t - 1] * h[v * HP + t - 1];
                if (t < TT - 1) acc += ar[t + 1] * h[v * HP + t + 1];
            }
            sA[v * HP + t] = acc;
        }
        __syncthreads();

        // ---- y[v][f] = (x1+mix) @ uw^T + ub  (M=V,K=T,N=T) -> sB ----
        // Hot GEMM: A non-trans + B trans => both fragments are single b64 loads.
        wmma_f32_gemm<VD, TT, TT, false, true, false>(sA, HP, wbuf, HP, sB, HP, ub);
        __syncthreads();

        // ---- LayerNorm over V per (b,f), then residual add ----
        for (int f = tid; f < TT; f += 256) {
            float m = 0.f;
            for (int v = 0; v < VD; ++v) m += sB[v * HP + f];
            m *= (1.0f / (float)VD);
            float var = 0.f;
            for (int v = 0; v < VD; ++v) {
                const float d = sB[v * HP + f] - m;
                var += d * d;
            }
            var *= (1.0f / (float)VD);
            mu[f]   = m;
            rstd[f] = rsqrtf(var + 1e-5f);
        }
        __syncthreads();
        for (int i = tid; i < VD * TT; i += 256) {
            const int v = i / TT, f = i % TT;
            h[v * HP + f] += (sB[v * HP + f] - mu[f]) * rstd[f] * la[v] + lb[v];
        }
        __syncthreads();
    }

    // ---- g[v][t] = h @ out_weight  (M=V,K=T,N=T) -> sA ----
    wmma_f32_gemm<VD, TT, TT, false, false, false>(h, HP, out_w, TT, sA, HP, nullptr);
    __syncthreads();

    // ---- out[b][t][v] = g^T @ fc_out_w^T + b  (M=T,K=V,N=V), direct to HBM ----
    wmma_f32_gemm<TT, VD, VD, true, true, true>(sA, HP, fc_out_w, VD,
                                               out + (size_t)b * TT * VD, VD, fc_out_b);
}

extern "C" void kernel_launch(void* const* d_in, const int* in_sizes, int n_in,
                              void* d_out, int out_size, void* d_ws, size_t ws_size,
                              hipStream_t stream) {
    (void)n_in; (void)out_size; (void)d_ws; (void)ws_size;
    const float* x        = (const float*)d_in[0];
    const float* fc_in_w  = (const float*)d_in[1];
    const float* fc_in_b  = (const float*)d_in[2];
    const float* in_w     = (const float*)d_in[3];
    const float* adj_j    = (const float*)d_in[4];
    const float* adj_t    = (const float*)d_in[5];
    const float* adj_jc   = (const float*)d_in[6];
    const float* adj_tj   = (const float*)d_in[7];
    const float* upd_w    = (const float*)d_in[8];
    const float* upd_b    = (const float*)d_in[9];
    const float* ln_a     = (const float*)d_in[10];
    const float* ln_b     = (const float*)d_in[11];
    const float* mlp      = (const float*)d_in[12];
    const float* out_w    = (const float*)d_in[13];
    const float* fc_out_w = (const float*)d_in[14];
    const float* fc_out_b = (const float*)d_in[15];
    const float* gumbel   = (const float*)d_in[16];
    // d_in[17] = edge_index (unused by the reference forward)

    const int Bn = in_sizes[0] / (TT * VD);   // batch from x's flat size
    gcnext_fused<<<Bn, 256, 0, stream>>>(
        x, fc_in_w, fc_in_b, in_w, adj_j, adj_t, adj_jc, adj_tj,
        upd_w, upd_b, ln_a, ln_b, mlp, out_w, fc_out_w, fc_out_b,
        gumbel, (float*)d_out, Bn);
}